// GATLayerOut_1932735283945
// MI455X (gfx1250) — compile-verified
//
#include <hip/hip_runtime.h>
#include <hip/hip_bf16.h>

typedef __attribute__((ext_vector_type(2))) float v2f;
typedef __attribute__((ext_vector_type(8))) float v8f;

#define NEG_INF_F (-9e15f)
#define ATT_STRIDE 260   // dword row stride: A-operand b64 LDS reads hit all 64 banks once

// ---------------------------------------------------------------------------
// Kernel 1: h = input @ W   (per batch: 256x64 = 256x64 @ 64x64)
// grid = (16 row-tiles, 16 batches), block = 128 (4 waves, one 16-col tile each)
// ---------------------------------------------------------------------------
__global__ __launch_bounds__(128) void k_h(const float* __restrict__ inp,
                                           const float* __restrict__ W,
                                           float* __restrict__ h) {
    const int b  = blockIdx.y;
    const int i0 = blockIdx.x * 16;
    const int lane = threadIdx.x & 31;
    const int wv   = threadIdx.x >> 5;
    const int n0   = wv * 16;
    const int m  = lane & 15;   // row (A) / col (B,C) within tile
    const int p  = lane >> 4;   // K-pair selector

    // A: input[b, i0+m, k0 + 2p + v]   (v = vgpr 0/1, contiguous -> v2f load)
    const float* A0 = inp + (size_t)b * 16384 + (size_t)(i0 + m) * 64 + 2 * p;
    // B: W[k0 + 2p + v, n0 + m]
    const float* B0 = W + (size_t)(2 * p) * 64 + n0 + m;

    v8f acc = {};
#pragma unroll
    for (int k0 = 0; k0 < 64; k0 += 4) {
        v2f av = *(const v2f*)(A0 + k0);
        v2f bv;
        bv.x = B0[k0 * 64];
        bv.y = B0[k0 * 64 + 64];
        acc = __builtin_amdgcn_wmma_f32_16x16x4_f32(false, av, false, bv,
                                                    (short)0, acc, false, false);
    }

    // D layout: vgpr d -> row d + 8p, col m
    float* H0 = h + (size_t)b * 16384 + (size_t)(i0 + p * 8) * 64 + n0 + m;
#pragma unroll
    for (int d = 0; d < 8; ++d) H0[d * 64] = acc[d];
}

// ---------------------------------------------------------------------------
// Kernel 2: s1[r] = h[r] . a[0:64], s2[r] = h[r] . a[64:128]   (r = 0..B*N-1)
// ---------------------------------------------------------------------------
__global__ __launch_bounds__(256) void k_s(const float* __restrict__ h,
                                           const float* __restrict__ avec,
                                           float* __restrict__ s1,
                                           float* __restrict__ s2) {
    const int r = blockIdx.x * 256 + threadIdx.x;   // 0..4095
    const float* hr = h + (size_t)r * 64;
    float a1 = 0.f, a2 = 0.f;
#pragma unroll 8
    for (int f = 0; f < 64; ++f) {
        float v = hr[f];
        a1 += v * avec[f];
        a2 += v * avec[64 + f];
    }
    s1[r] = a1;
    s2[r] = a2;
}

// ---------------------------------------------------------------------------
// Kernel 3: e -> mask -> softmax -> att @ h -> elu
// grid = (16 row-tiles, 16 batches), block = 128 (4 waves)
// ---------------------------------------------------------------------------
__global__ __launch_bounds__(128) void k_att(const int* __restrict__ adj,
                                             const float* __restrict__ h,
                                             const float* __restrict__ s1,
                                             const float* __restrict__ s2,
                                             float* __restrict__ out) {
    __shared__ __align__(16) float att[16 * ATT_STRIDE]; // 16 x 256 scores, padded rows
    __shared__ float s1l[256];
    __shared__ float s2l[256];
    __shared__ float red[16 * 8];
    __shared__ float rmax[16];
    __shared__ float rinv[16];

    const int b   = blockIdx.y;
    const int i0  = blockIdx.x * 16;
    const int tid = threadIdx.x;

    const int* adjp = adj + (size_t)b * 65536;

    // warm the 16KB adj tile for this block (global_prefetch_b8): 128 lines of 128B
    __builtin_prefetch(&adjp[i0 * 256 + tid * 32], 0, 0);

    // stage s1/s2 for this batch
    for (int r = tid; r < 256; r += 128) {
        s1l[r] = s1[b * 256 + r];
        s2l[r] = s2[b * 256 + r];
    }
    __syncthreads();

    // ---- build masked leaky-relu scores (exploits the reference's interleave) ----
    const bool lowblock = (i0 < 128);   // uniform per block (i0 multiple of 16)
    for (int idx = tid; idx < 4096; idx += 128) {
        const int ii = idx >> 8;
        const int j  = idx & 255;
        const int i  = i0 + ii;
        int r1, r2;
        if (lowblock) { r1 = 2 * i + (j >> 7); r2 = r1; }
        else          { r1 = (2 * j) & 255;    r2 = (2 * j + 1) & 255; }
        float ev = s1l[r1] + s2l[r2];
        ev = ev > 0.f ? ev : 0.01f * ev;                       // leaky_relu(0.01)
        att[ii * ATT_STRIDE + j] = (adjp[i * 256 + j] > 0) ? ev : NEG_INF_F;
    }
    __syncthreads();

    // ---- softmax stats: 8 threads per row ----
    const int ii  = tid >> 3;
    const int sub = tid & 7;
    {
        float mx = -3.4e38f;
        const int jb = sub * 32;
        for (int j = jb; j < jb + 32; ++j) mx = fmaxf(mx, att[ii * ATT_STRIDE + j]);
        red[ii * 8 + sub] = mx;
    }
    __syncthreads();
    if (sub == 0) {
        float m0 = red[ii * 8];
        for (int q = 1; q < 8; ++q) m0 = fmaxf(m0, red[ii * 8 + q]);
        rmax[ii] = m0;
    }
    __syncthreads();
    {
        const float rm = rmax[ii];
        float sm = 0.f;
        const int jb = sub * 32;
        for (int j = jb; j < jb + 32; ++j) {
            float p = __expf(att[ii * ATT_STRIDE + j] - rm);
            att[ii * ATT_STRIDE + j] = p;
            sm += p;
        }
        red[ii * 8 + sub] = sm;
    }
    __syncthreads();
    if (sub == 0) {
        float s = 0.f;
        for (int q = 0; q < 8; ++q) s += red[ii * 8 + q];
        rinv[ii] = 1.0f / s;   // fold normalization into the GEMM epilogue
    }
    __syncthreads();

    // ---- att(16x256) @ h(256x64) via V_WMMA_F32_16X16X4_F32 ----
    const int lane = tid & 31;
    const int wv   = tid >> 5;
    const int n0   = wv * 16;
    const int m  = lane & 15;
    const int p  = lane >> 4;

    // A operand: aligned b64 LDS reads, conflict-free with ATT_STRIDE=260
    const float* Arow = att + m * ATT_STRIDE + 2 * p;
    // B operand: h[b, k0+2p+v, n0+m] from global (each element read once per block)
    const float* H0   = h + (size_t)b * 16384 + (size_t)(2 * p) * 64 + n0 + m;

    v8f acc = {};
#pragma unroll 8
    for (int k0 = 0; k0 < 256; k0 += 4) {
        v2f av = *(const v2f*)(Arow + k0);
        v2f bv;
        bv.x = H0[k0 * 64];
        bv.y = H0[k0 * 64 + 64];
        acc = __builtin_amdgcn_wmma_f32_16x16x4_f32(false, av, false, bv,
                                                    (short)0, acc, false, false);
    }

    // epilogue: scale by 1/rowsum, elu, store
    float* O = out + (size_t)b * 16384 + (size_t)(i0 + p * 8) * 64 + n0 + m;
#pragma unroll
    for (int d = 0; d < 8; ++d) {
        float x = acc[d] * rinv[d + p * 8];
        x = x > 0.f ? x : (__expf(x) - 1.0f);   // elu, alpha=1
        O[d * 64] = x;
    }
}

// ---------------------------------------------------------------------------
extern "C" void kernel_launch(void* const* d_in, const int* in_sizes, int n_in,
                              void* d_out, int out_size, void* d_ws, size_t ws_size,
                              hipStream_t stream) {
    const float* inp  = (const float*)d_in[0];   // (16,256,64) f32
    const int*   adj  = (const int*)d_in[1];     // (16,256,256) i32
    const float* W    = (const float*)d_in[2];   // (64,64) f32
    const float* avec = (const float*)d_in[3];   // (128,) f32
    float* out = (float*)d_out;                  // (16,256,64) f32

    float* hws = (float*)d_ws;                   // 16*256*64 floats
    float* s1  = hws + 16 * 256 * 64;            // 4096 floats
    float* s2  = s1 + 4096;                      // 4096 floats

    dim3 gridT(16, 16);
    k_h<<<gridT, 128, 0, stream>>>(inp, W, hws);
    k_s<<<16, 256, 0, stream>>>(hws, avec, s1, s2);
    k_att<<<gridT, 128, 0, stream>>>(adj, hws, s1, s2, out);
}